// ChromDropout_52080773431624
// MI455X (gfx1250) — compile-verified
//
#include <hip/hip_runtime.h>
#include <stdint.h>

// ---------------- problem constants (from reference) ----------------
#define B_ROWS      16384
#define G_GENES     19064
#define NUM_CHROMS  23
#define N_DROP      4
#define NV          (G_GENES / 4)     // 4766 float4 chunks per row (19064 % 4 == 0)
#define ROWS_PB     16                // rows processed per block (gene tile reuse)
#define SCALE_KEEP  5.75f             // 1/RATIO = 23/4

typedef __attribute__((ext_vector_type(4))) float f4;
typedef __attribute__((ext_vector_type(4))) int   i4;

// Builtin expects: (v4i addrspace(1)*, v4i addrspace(3)*, imm offset, imm cpol)
typedef __attribute__((address_space(1))) i4 gi4;   // global
typedef __attribute__((address_space(3))) i4 li4;   // LDS

// Detect CDNA5 async global->LDS builtins (gfx1250). Fallback: plain register load.
#if defined(__has_builtin)
#if __has_builtin(__builtin_amdgcn_global_load_async_to_lds_b128) && \
    __has_builtin(__builtin_amdgcn_s_wait_asynccnt)
#define USE_ASYNC_LDS 1
#endif
#endif

// ---------------- kernel 1: per-row 23-bit drop mask ----------------
__device__ __forceinline__ uint32_t pcg_next(uint32_t& s) {
    s = s * 747796405u + 2891336453u;
    uint32_t w = ((s >> ((s >> 28) + 4u)) ^ s) * 277803737u;
    return (w >> 22) ^ w;
}

__global__ __launch_bounds__(256) void chrom_mask_kernel(uint32_t* __restrict__ masks) {
    int row = blockIdx.x * 256 + threadIdx.x;
    if (row >= B_ROWS) return;
    // deterministic per-row stream (stands in for jax.random.permutation(key(42)))
    uint32_t s = (uint32_t)row * 0x9E3779B9u + 0x85EBCA6Bu;
    (void)pcg_next(s); (void)pcg_next(s);
    uint32_t mask = 0u;
#pragma unroll
    for (int i = 0; i < N_DROP; ++i) {
        uint32_t r = pcg_next(s) % (uint32_t)(NUM_CHROMS - i);
        int c = 0;
        for (;;) {                      // find r-th zero bit -> distinct picks
            if (!((mask >> c) & 1u)) { if (r == 0u) break; --r; }
            ++c;
        }
        mask |= (1u << c);
    }
    masks[row] = mask;                  // bit c set == chromosome c dropped
}

// ---------------- kernel 2: streaming apply ----------------
// Grid: x = gene tiles (256 float4 each), y = row groups of ROWS_PB.
// chrom_ids tile (4 KB) staged once per block via CDNA5 async copy to LDS,
// then held in registers across the 16-row loop. x/out stream with NT hints.
__global__ __launch_bounds__(256) void chrom_apply_kernel(
    const float* __restrict__ x,
    const int*   __restrict__ chrom_ids,
    const uint32_t* __restrict__ masks,
    float* __restrict__ out)
{
    __shared__ int lds_ids[1024];

    const int tid     = threadIdx.x;
    const int v       = blockIdx.x * 256 + tid;     // float4 index within row
    const int rowbase = blockIdx.y * ROWS_PB;
    const bool inb    = (v < NV);

    i4 cid;
#ifdef USE_ASYNC_LDS
    if (inb) {
        __builtin_amdgcn_global_load_async_to_lds_b128(
            (gi4*)(chrom_ids + 4 * (size_t)v),
            (li4*)(&lds_ids[4 * tid]),
            /*offset=*/0, /*cpol=*/0);
    }
    __builtin_amdgcn_s_wait_asynccnt(0);            // own lane's 16B only -> wave wait suffices
    __syncthreads();
    cid = inb ? *(const i4*)(&lds_ids[4 * tid]) : (i4){0, 0, 0, 0};
#else
    cid = inb ? *(const i4*)(chrom_ids + 4 * (size_t)v) : (i4){0, 0, 0, 0};
#endif

#pragma unroll 4
    for (int r = 0; r < ROWS_PB; ++r) {
        const int row = rowbase + r;
        const uint32_t keep = ~masks[row];          // uniform -> s_load, bit=1 means keep
        if (inb) {
            const f4* px = (const f4*)(x   + (size_t)row * G_GENES) + v;
            f4*       po = (f4*)      (out + (size_t)row * G_GENES) + v;
            f4 xv = __builtin_nontemporal_load(px); // stream: don't pollute 192MB L2
            f4 o;
            o.x = xv.x * (((keep >> cid.x) & 1u) ? SCALE_KEEP : 0.0f);
            o.y = xv.y * (((keep >> cid.y) & 1u) ? SCALE_KEEP : 0.0f);
            o.z = xv.z * (((keep >> cid.z) & 1u) ? SCALE_KEEP : 0.0f);
            o.w = xv.w * (((keep >> cid.w) & 1u) ? SCALE_KEEP : 0.0f);
            __builtin_nontemporal_store(o, po);
        }
    }
}

// ---------------- launcher ----------------
extern "C" void kernel_launch(void* const* d_in, const int* in_sizes, int n_in,
                              void* d_out, int out_size, void* d_ws, size_t ws_size,
                              hipStream_t stream) {
    const float*    x     = (const float*)d_in[0];   // [B, G] f32
    const int*      chrom = (const int*)d_in[1];     // [G] i32
    float*          out   = (float*)d_out;           // [B, G] f32
    uint32_t*       masks = (uint32_t*)d_ws;         // 64 KB scratch

    chrom_mask_kernel<<<(B_ROWS + 255) / 256, 256, 0, stream>>>(masks);

    dim3 grid((NV + 255) / 256, B_ROWS / ROWS_PB);   // (19, 1024)
    chrom_apply_kernel<<<grid, dim3(256, 1, 1), 0, stream>>>(x, chrom, masks, out);
}